// GCN_1657857376632
// MI455X (gfx1250) — compile-verified
//
#include <hip/hip_runtime.h>
#include <hip/hip_bf16.h>

typedef float v2f __attribute__((ext_vector_type(2)));
typedef float v8f __attribute__((ext_vector_type(8)));

// ---------------- init / utility ----------------
__global__ void k_fill_f32(float* __restrict__ p, float v, int n) {
  int t = blockIdx.x * blockDim.x + threadIdx.x;
  if (t < n) p[t] = v;
}
__global__ void k_fill_u32(unsigned* __restrict__ p, unsigned v, int n) {
  int t = blockIdx.x * blockDim.x + threadIdx.x;
  if (t < n) p[t] = v;
}

// deg[dst[e]] += w[e]   (deg pre-filled with 1.0 = self-loop weight)
__global__ void k_edge_degree(const int* __restrict__ dst, const float* __restrict__ ew,
                              float* __restrict__ deg, int E) {
  int e = blockIdx.x * blockDim.x + threadIdx.x;
  if (e < E) atomicAdd(&deg[dst[e]], ew[e]);
}

__global__ void k_deg_to_dinv(float* __restrict__ deg, int n) {
  int i = blockIdx.x * blockDim.x + threadIdx.x;
  if (i < n) {
    float d = deg[i];
    deg[i] = d > 0.f ? rsqrtf(d) : 0.f;
  }
}

__global__ void k_edge_norm(const int* __restrict__ src, const int* __restrict__ dst,
                            const float* __restrict__ ew, const float* __restrict__ dinv,
                            float* __restrict__ norm, int E) {
  int e = blockIdx.x * blockDim.x + threadIdx.x;
  if (e < E) norm[e] = dinv[src[e]] * ew[e] * dinv[dst[e]];
}

// ---------------- WMMA GEMMs (fp32, 16x16x4) ----------------
static __device__ __forceinline__ v8f wmma_f32_k4(v2f a, v2f b, v8f c) {
  return __builtin_amdgcn_wmma_f32_16x16x4_f32(false, a, false, b, (short)0, c,
                                               false, false);
}

// OUT[N,64] = X[N,256] @ W[256,64].  128 threads = 4 waves, 1 wave = 16-row tile.
__global__ __launch_bounds__(128)
void k_gemm1_wmma(const float* __restrict__ X, const float* __restrict__ W,
                  float* __restrict__ OUT, int nTiles) {
  __shared__ float sW[256 * 64];
  for (int i = threadIdx.x; i < 256 * 64; i += 128) sW[i] = W[i];
  __syncthreads();
  const int wave = threadIdx.x >> 5;
  const int lane = threadIdx.x & 31;
  const int tile = blockIdx.x * 4 + wave;
  if (tile >= nTiles) return;               // wave-uniform: EXEC stays all-1s
  const int m0   = tile << 4;
  const int half = lane >> 4;               // 0: k,k+1   1: k+2,k+3
  const int l16  = lane & 15;
  const float* __restrict__ xrow = X + (size_t)(m0 + l16) * 256;
  v8f acc0 = {}, acc1 = {}, acc2 = {}, acc3 = {};
  for (int k = 0; k < 256; k += 4) {
    v2f a;
    a.x = xrow[k + half * 2 + 0];
    a.y = xrow[k + half * 2 + 1];
    const float* wr0 = &sW[(k + half * 2 + 0) * 64 + l16];
    const float* wr1 = &sW[(k + half * 2 + 1) * 64 + l16];
    v2f b0; b0.x = wr0[0];  b0.y = wr1[0];
    v2f b1; b1.x = wr0[16]; b1.y = wr1[16];
    v2f b2; b2.x = wr0[32]; b2.y = wr1[32];
    v2f b3; b3.x = wr0[48]; b3.y = wr1[48];
    acc0 = wmma_f32_k4(a, b0, acc0);
    acc1 = wmma_f32_k4(a, b1, acc1);
    acc2 = wmma_f32_k4(a, b2, acc2);
    acc3 = wmma_f32_k4(a, b3, acc3);
  }
  float* orow = OUT + (size_t)(m0 + half * 8) * 64 + l16;
#pragma unroll
  for (int v = 0; v < 8; ++v) {
    orow[(size_t)v * 64 +  0] = acc0[v];
    orow[(size_t)v * 64 + 16] = acc1[v];
    orow[(size_t)v * 64 + 32] = acc2[v];
    orow[(size_t)v * 64 + 48] = acc3[v];
  }
}

// OUT[N,32] = H[N,64] @ W[64,32]
__global__ __launch_bounds__(128)
void k_gemm2_wmma(const float* __restrict__ H, const float* __restrict__ W,
                  float* __restrict__ OUT, int nTiles) {
  __shared__ float sW[64 * 32];
  for (int i = threadIdx.x; i < 64 * 32; i += 128) sW[i] = W[i];
  __syncthreads();
  const int wave = threadIdx.x >> 5;
  const int lane = threadIdx.x & 31;
  const int tile = blockIdx.x * 4 + wave;
  if (tile >= nTiles) return;
  const int m0   = tile << 4;
  const int half = lane >> 4;
  const int l16  = lane & 15;
  const float* __restrict__ xrow = H + (size_t)(m0 + l16) * 64;
  v8f acc0 = {}, acc1 = {};
  for (int k = 0; k < 64; k += 4) {
    v2f a;
    a.x = xrow[k + half * 2 + 0];
    a.y = xrow[k + half * 2 + 1];
    const float* wr0 = &sW[(k + half * 2 + 0) * 32 + l16];
    const float* wr1 = &sW[(k + half * 2 + 1) * 32 + l16];
    v2f b0; b0.x = wr0[0];  b0.y = wr1[0];
    v2f b1; b1.x = wr0[16]; b1.y = wr1[16];
    acc0 = wmma_f32_k4(a, b0, acc0);
    acc1 = wmma_f32_k4(a, b1, acc1);
  }
  float* orow = OUT + (size_t)(m0 + half * 8) * 32 + l16;
#pragma unroll
  for (int v = 0; v < 8; ++v) {
    orow[(size_t)v * 32 +  0] = acc0[v];
    orow[(size_t)v * 32 + 16] = acc1[v];
  }
}

// ---------------- edge scatter: AGG[dst] += norm * H[src] ----------------
template <int F, int SHIFT>  // F feats, CH = F/4 chunks, SHIFT = log2(CH)
__global__ void k_scatter(const int* __restrict__ src, const int* __restrict__ dst,
                          const float* __restrict__ norm, const float* __restrict__ H,
                          float* __restrict__ AGG, int nwork) {
  int tid = blockIdx.x * blockDim.x + threadIdx.x;
  if (tid >= nwork) return;
  const int CH = F / 4;
  int c = tid & (CH - 1);
  int e = tid >> SHIFT;
  int s = src[e], d = dst[e];
  float nm = norm[e];
  const float4* hp = (const float4*)(H + (size_t)s * F);
  float4 v = hp[c];
  float* out = AGG + (size_t)d * F + (c << 2);
  atomicAdd(out + 0, v.x * nm);
  atomicAdd(out + 1, v.y * nm);
  atomicAdd(out + 2, v.z * nm);
  atomicAdd(out + 3, v.w * nm);
}

// h = relu(AGG + dinv^2 * HPRE + bias)   (self-loop folded in; in-place on AGG)
template <int F, int SHIFT>
__global__ void k_combine(float* __restrict__ AGG, const float* __restrict__ HPRE,
                          const float* __restrict__ dinv, const float* __restrict__ bias,
                          int total) {
  int tid = blockIdx.x * blockDim.x + threadIdx.x;
  if (tid >= total) return;
  int i = tid >> SHIFT;
  int f = tid & (F - 1);
  float di = dinv[i];
  float v = AGG[tid] + di * di * HPRE[tid] + bias[f];
  AGG[tid] = v > 0.f ? v : 0.f;
}

// segment max-pool: int atomicMax on float bits (valid: values >= 0, init = -inf bits)
__global__ void k_pool_max(const float* __restrict__ H2, const int* __restrict__ batch,
                           int* __restrict__ pooled, int total) {
  int tid = blockIdx.x * blockDim.x + threadIdx.x;
  if (tid >= total) return;
  int i = tid >> 5;
  int f = tid & 31;
  atomicMax(&pooled[batch[i] * 32 + f], __float_as_int(H2[tid]));
}

__global__ void k_final(const int* __restrict__ pooled, const float* __restrict__ Wlin,
                        const float* __restrict__ blin, float* __restrict__ out, int G) {
  int t = blockIdx.x * blockDim.x + threadIdx.x;
  if (t >= G * 4) return;
  int g = t >> 2, o = t & 3;
  const int* pb = pooled + g * 32;
  float acc = blin[o];
#pragma unroll
  for (int j = 0; j < 32; ++j) acc += __int_as_float(pb[j]) * Wlin[j * 4 + o];
  out[t] = acc;
}

// ---------------- launcher ----------------
extern "C" void kernel_launch(void* const* d_in, const int* in_sizes, int n_in,
                              void* d_out, int out_size, void* d_ws, size_t ws_size,
                              hipStream_t stream) {
  const float* x     = (const float*)d_in[0];
  const int*   ei    = (const int*)d_in[1];
  const float* ew    = (const float*)d_in[2];
  const int*   batch = (const int*)d_in[3];
  const float* W1    = (const float*)d_in[4];
  const float* b1    = (const float*)d_in[5];
  const float* W2    = (const float*)d_in[6];
  const float* b2    = (const float*)d_in[7];
  const float* Wlin  = (const float*)d_in[8];
  const float* blin  = (const float*)d_in[9];

  const int N = in_sizes[0] / 256;   // 100000 (multiple of 16)
  const int E = in_sizes[2];         // 3200000
  const int G = out_size / 4;        // 64
  const int* src = ei;
  const int* dst = ei + E;

  char*  ws  = (char*)d_ws;
  size_t off = 0;
  auto carve = [&](size_t bytes) {
    size_t o = off;
    off += (bytes + 255) & ~(size_t)255;
    return o;
  };
  float* dinv   = (float*)(ws + carve((size_t)N * 4));
  float* norm   = (float*)(ws + carve((size_t)E * 4));
  float* h1pre  = (float*)(ws + carve((size_t)N * 64 * 4));
  float* agg1   = (float*)(ws + carve((size_t)N * 64 * 4));  // becomes h1 in-place
  float* h2pre  = (float*)(ws + carve((size_t)N * 32 * 4));
  float* agg2   = (float*)(ws + carve((size_t)N * 32 * 4));  // becomes h2 in-place
  int*   pooled = (int*)  (ws + carve((size_t)G * 32 * 4));
  (void)ws_size; (void)n_in;

  const int TPB = 256;
  auto nb = [&](int n) { return (n + TPB - 1) / TPB; };

  // degree (self-loop weight 1.0 baked into init) -> dinv
  k_fill_f32<<<nb(N), TPB, 0, stream>>>(dinv, 1.0f, N);
  k_edge_degree<<<nb(E), TPB, 0, stream>>>(dst, ew, dinv, E);
  k_deg_to_dinv<<<nb(N), TPB, 0, stream>>>(dinv, N);
  k_edge_norm<<<nb(E), TPB, 0, stream>>>(src, dst, ew, dinv, norm, E);

  // layer 1
  const int nTiles = N / 16;
  k_gemm1_wmma<<<(nTiles + 3) / 4, 128, 0, stream>>>(x, W1, h1pre, nTiles);
  k_fill_f32<<<nb(N * 64), TPB, 0, stream>>>(agg1, 0.0f, N * 64);
  k_scatter<64, 4><<<nb(E * 16), TPB, 0, stream>>>(src, dst, norm, h1pre, agg1, E * 16);
  k_combine<64, 6><<<nb(N * 64), TPB, 0, stream>>>(agg1, h1pre, dinv, b1, N * 64);

  // layer 2
  k_gemm2_wmma<<<(nTiles + 3) / 4, 128, 0, stream>>>(agg1, W2, h2pre, nTiles);
  k_fill_f32<<<nb(N * 32), TPB, 0, stream>>>(agg2, 0.0f, N * 32);
  k_scatter<32, 3><<<nb(E * 8), TPB, 0, stream>>>(src, dst, norm, h2pre, agg2, E * 8);
  k_combine<32, 5><<<nb(N * 32), TPB, 0, stream>>>(agg2, h2pre, dinv, b2, N * 32);

  // pool + head
  k_fill_u32<<<nb(G * 32), TPB, 0, stream>>>((unsigned*)pooled, 0xFF800000u, G * 32);
  k_pool_max<<<nb(N * 32), TPB, 0, stream>>>(agg2, batch, pooled, N * 32);
  k_final<<<nb(G * 4), TPB, 0, stream>>>(pooled, Wlin, blin, (float*)d_out, G);
}